// SpikingLayer_45380624450077
// MI455X (gfx1250) — compile-verified
//
#include <hip/hip_runtime.h>

// SNN layer fused kernel for MI455X (gfx1250, wave32).
// h = x @ w  via V_WMMA_F32_16X16X4_F32 into LDS, then in-block scan over T.

#define BDIM 128
#define TDIM 200
#define IDIM 700
#define ODIM 1024

#define NTILE 64            // O-columns per workgroup
#define MROWS 208           // T padded to 13 * 16
#define MT    13            // number of 16-row M tiles
#define KBLK  16            // K per staging block (4 wmma k-steps)

#define XS_STRIDE 18        // pad 16 -> 18 (even, conflict-free b64 A reads)
#define WS_STRIDE 72        // pad 64 -> 72 (half-wave K rows hit disjoint banks)
#define HS_STRIDE 66        // pad 64 -> 66 (half-wave C stores hit disjoint banks)

typedef __attribute__((ext_vector_type(2))) float v2f;
typedef __attribute__((ext_vector_type(8))) float v8f;

__launch_bounds__(256)
__global__ void snn_fused_kernel(const float* __restrict__ x,
                                 const float* __restrict__ w,
                                 const float* __restrict__ alpha,
                                 const float* __restrict__ beta,
                                 const float* __restrict__ th,
                                 float* __restrict__ out) {
    __shared__ float xs[MROWS * XS_STRIDE];   // 14,976 B
    __shared__ float ws[KBLK  * WS_STRIDE];   //  4,608 B
    __shared__ float hs[MROWS * HS_STRIDE];   // 54,912 B

    const int tid    = threadIdx.x;
    const int b      = blockIdx.y;
    const int o0     = blockIdx.x * NTILE;

    const int lane   = tid & 31;       // wave32
    const int wv     = tid >> 5;       // 8 waves per block
    const int lane16 = lane & 15;
    const int half   = lane >> 4;      // 0: lanes 0-15, 1: lanes 16-31
    const int koff   = half * 2;       // A/B K offset per half-wave (16x16x4 layout)

    v8f acc[2][4];
    const v8f vzero = {0.f,0.f,0.f,0.f,0.f,0.f,0.f,0.f};
#pragma unroll
    for (int r = 0; r < 2; ++r)
#pragma unroll
        for (int n = 0; n < 4; ++n) acc[r][n] = vzero;

    for (int k0 = 0; k0 < IDIM; k0 += KBLK) {
        // ---- stage x tile: rows = time (padded to 208), cols = K block ----
        for (int task = tid; task < MROWS * 4; task += 256) {
            const int r  = task >> 2;
            const int cg = task & 3;
            const int k  = k0 + cg * 4;
            float4 v = make_float4(0.f, 0.f, 0.f, 0.f);
            if (r < TDIM) {
                const float* src = x + ((size_t)b * TDIM + r) * IDIM + k;
                if (k + 3 < IDIM) {
                    v = *(const float4*)src;
                } else {
                    if (k + 0 < IDIM) v.x = src[0];
                    if (k + 1 < IDIM) v.y = src[1];
                    if (k + 2 < IDIM) v.z = src[2];
                    if (k + 3 < IDIM) v.w = src[3];
                }
            }
            float* dst = &xs[r * XS_STRIDE + cg * 4];
            ((v2f*)dst)[0] = (v2f){v.x, v.y};
            ((v2f*)dst)[1] = (v2f){v.z, v.w};
        }
        // ---- stage w tile: [KBLK][64], one float4 per thread ----
        {
            const int kk = tid >> 4;
            const int c4 = (tid & 15) << 2;
            const int k  = k0 + kk;
            float4 v = make_float4(0.f, 0.f, 0.f, 0.f);
            if (k < IDIM) v = *(const float4*)(w + (size_t)k * ODIM + o0 + c4);
            *(float4*)&ws[kk * WS_STRIDE + c4] = v;
        }
        __syncthreads();

        // ---- 4 wmma k-steps of K=4 each ----
#pragma unroll
        for (int kk = 0; kk < 4; ++kk) {
            const int kl = kk * 4 + koff;
            v2f bf[4];
#pragma unroll
            for (int nb = 0; nb < 4; ++nb) {
                bf[nb].x = ws[(kl    ) * WS_STRIDE + nb * 16 + lane16];
                bf[nb].y = ws[(kl + 1) * WS_STRIDE + nb * 16 + lane16];
            }
#pragma unroll
            for (int rep = 0; rep < 2; ++rep) {
                const int mt = wv + rep * 8;     // wave-uniform branch: EXEC stays all-1s
                if (mt < MT) {
                    const int r = mt * 16 + lane16;
                    const v2f af = *(const v2f*)&xs[r * XS_STRIDE + kl];
#pragma unroll
                    for (int nb = 0; nb < 4; ++nb) {
                        acc[rep][nb] = __builtin_amdgcn_wmma_f32_16x16x4_f32(
                            false, af, false, bf[nb], (short)0, acc[rep][nb],
                            false, false);
                    }
                }
            }
        }
        __syncthreads();
    }

    // ---- write accumulators (h tile) to LDS ----
#pragma unroll
    for (int rep = 0; rep < 2; ++rep) {
        const int mt = wv + rep * 8;
        if (mt < MT) {
#pragma unroll
            for (int nb = 0; nb < 4; ++nb) {
#pragma unroll
                for (int i = 0; i < 8; ++i) {
                    const int row = mt * 16 + half * 8 + i;   // C layout: M = i + 8*half
                    hs[row * HS_STRIDE + nb * 16 + lane16] = acc[rep][nb][i];
                }
            }
        }
    }
    __syncthreads();

    // ---- sequential scan over T (one thread per o-column) ----
    if (tid < NTILE) {
        const int o = o0 + tid;
        const float al = alpha[o];
        const float be = beta[o];
        const float tv = th[o];

        const size_t BTO = (size_t)BDIM * TDIM * ODIM;
        float* out_syn = out;
        float* out_mem = out + BTO;
        float* out_spk = out + 2 * BTO;

        const size_t idx0 = ((size_t)b * TDIM) * ODIM + o;
        out_syn[idx0] = 0.f;
        out_mem[idx0] = 0.f;
        out_spk[idx0] = 0.f;

        float syn = 0.f, mem = 0.f, ref = 0.f;
        for (int s = 0; s < TDIM - 1; ++s) {
            const float hv  = hs[s * HS_STRIDE + tid];
            const bool  c   = (mem - tv) > 0.f;
            const float spk = c ? 1.f : 0.f;
            if (c) ref = 5.0f;                                  // TREF, before gate
            const float gate = (ref < 1.f) ? 1.f : 0.f;
            const float nmem = (be * mem + (1.f - be) * syn - (c ? tv : 0.f)) * gate;
            const float nsyn = al * syn + hv;
            ref = (ref > 0.f) ? (ref - 1.f) : ref;
            syn = nsyn;
            mem = nmem;
            const size_t idx = idx0 + (size_t)(s + 1) * ODIM;
            out_syn[idx] = syn;
            out_mem[idx] = mem;
            out_spk[idx] = spk;
        }
    }
}

extern "C" void kernel_launch(void* const* d_in, const int* in_sizes, int n_in,
                              void* d_out, int out_size, void* d_ws, size_t ws_size,
                              hipStream_t stream) {
    const float* x     = (const float*)d_in[0];
    const float* w     = (const float*)d_in[1];
    const float* alpha = (const float*)d_in[2];
    const float* beta  = (const float*)d_in[3];
    const float* th    = (const float*)d_in[4];
    float* out = (float*)d_out;

    dim3 grid(ODIM / NTILE, BDIM);   // (16, 128)
    snn_fused_kernel<<<grid, 256, 0, stream>>>(x, w, alpha, beta, th, out);
}